// IoULovaszLoss_1623497638734
// MI455X (gfx1250) — compile-verified
//
#include <hip/hip_runtime.h>
#include <hip/hip_bf16.h>
#include <math.h>

// Problem constants (match reference)
#define BB 4
#define NN 131072        // 2^17
#define II 16
#define CCLS 4
#define NSEQ (BB*CCLS*II)   // 256 Lovasz sequences
#define CHUNK 2048

typedef __attribute__((ext_vector_type(2))) float v2f;
typedef __attribute__((ext_vector_type(8))) float v8f;

// Workspace layout (in 32-bit words)
#define CS_OFF 0          // centerSums: B*I*4 floats (sx,sy,sz,count)
#define FI_OFF 256        // firstIdx:   B*I ints
#define MH_OFF 320        // M table:    B*5*16 u32  (sem x instance counts)
#define CT_OFF 640        // centers:    B*I*4 floats (cx,cy,cz,count)
#define VA_OFF 896        // valid:      256 floats
#define NV_OFF 1152       // n_valid (float)
#define LS_OFF 1153       // loss accumulator (float)
#define HDR_WORDS 1280    // histogram starts here: NSEQ * nbins * 4 words
// per-bin 4 words: [0]=cntB(u32) [1]=cntF(u32) [2]=sumB(f32) [3]=sumF(f32)

__global__ void iou_init(unsigned* __restrict__ ws, int totalWords) {
  int idx = blockIdx.x * blockDim.x + threadIdx.x;
  if (idx < totalWords)
    ws[idx] = (idx >= FI_OFF && idx < FI_OFF + BB*II) ? 0x7FFFFFFFu : 0u;
}

// Phase 1: per-instance center sums via WMMA f32 16x16x4, plus
// sem-x-instance joint counts and first-occurrence index.
__global__ __launch_bounds__(256) void iou_stats(
    const float* __restrict__ pts, const int* __restrict__ tgt,
    const int* __restrict__ sem, const float* __restrict__ embd,
    const int* __restrict__ vrp, unsigned* __restrict__ ws) {
  const int b     = blockIdx.x;
  const int nbase = blockIdx.y * CHUNK;
  const int tid   = threadIdx.x;

  float*    centerSums = (float*)(ws + CS_OFF);
  int*      firstIdx   = (int*)(ws + FI_OFF);
  unsigned* Mh         = ws + MH_OFF;

  __shared__ unsigned Mloc[80];
  __shared__ int fminS[16];
  if (tid < 80) Mloc[tid] = 0u;
  if (tid < 16) fminS[tid] = 0x7FFFFFFF;
  __syncthreads();

  const int base = b * NN;
  for (int k = 0; k < CHUNK; k += 256) {
    const int n = nbase + k + tid;
    const int t = tgt[base + n];
    const int s = sem[base + n];
    atomicAdd(&Mloc[s * 16 + t], 1u);
    atomicMin(&fminS[t], n);
  }

  // WMMA: D[i][d] += mask[i,n] * embext[n,d], embext = (x,y,z,1,0...)
  // A (16x4 f32): lane holds rows M=lane&15, K = {0,1} (lanes 0-15) / {2,3} (16-31)
  // B (4x16 f32): mirrored layout, col = lane&15
  const float vr  = (float)vrp[0];
  const int lane  = tid & 31;
  const int row   = lane & 15;
  const int hi    = lane >> 4;
  const int wave  = tid >> 5;

  // Branchless B construction: always load a clamped (in-bounds) column,
  // then blend with lane-invariant constants -> no exec-mask churn in loop.
  const int   colc  = (row < 3) ? row : 2;   // safe column to load
  const float cKeep = (row < 3) ? 1.f : 0.f; // keep loaded value (cols 0..2)
  const float cOne  = (row == 3) ? 1.f : 0.f;// count column (col 3)

  v8f acc = {0.f, 0.f, 0.f, 0.f, 0.f, 0.f, 0.f, 0.f};
  const int wbase = nbase + wave * 256;      // 8 waves x 256 points = CHUNK
  for (int it = 0; it < 64; ++it) {
    const int n0 = wbase + it * 4;
    const int kk = hi * 2;
    const int t0 = tgt[base + n0 + kk];
    const int t1 = tgt[base + n0 + kk + 1];
    v2f a;
    a.x = (t0 == row) ? 1.f : 0.f;
    a.y = (t1 == row) ? 1.f : 0.f;
    const int i0 = (base + n0 + kk) * 3 + colc;
    const int i1 = i0 + 3;
    const float e0 = embd[i0] + pts[i0] * vr;
    const float e1 = embd[i1] + pts[i1] * vr;
    v2f bv;
    bv.x = e0 * cKeep + cOne;
    bv.y = e1 * cKeep + cOne;
    // stay ~64 points ahead of the stream (global_prefetch_b8)
    __builtin_prefetch(&tgt[base + n0 + kk + 64], 0, 1);
    __builtin_prefetch(&embd[i0 + 3 * 64], 0, 1);
    __builtin_prefetch(&pts[i0 + 3 * 64], 0, 1);
    acc = __builtin_amdgcn_wmma_f32_16x16x4_f32(false, a, false, bv,
                                                (short)0, acc, false, false);
  }
  // C/D layout: lane (col=lane&15) in VGPR v holds D[M = v + 8*hi][N = col]
  if (row < 4) {
#pragma unroll
    for (int v = 0; v < 8; ++v) {
      const int m = v + hi * 8;
      atomicAdd(&centerSums[(b * II + m) * 4 + row], acc[v]);
    }
  }
  __syncthreads();
  if (tid < 80) atomicAdd(&Mh[b * 80 + tid], Mloc[tid]);
  if (tid < 16 && fminS[tid] != 0x7FFFFFFF)
    atomicMin(&firstIdx[b * II + tid], fminS[tid]);
}

// Phase 2: centers, validity mask, bulk (exact 0 / exact 1) histogram seed.
__global__ __launch_bounds__(256) void iou_finalize(
    const int* __restrict__ sem, unsigned* __restrict__ ws, int nbins) {
  const int tid = threadIdx.x;
  float*    centerSums = (float*)(ws + CS_OFF);
  int*      firstIdx   = (int*)(ws + FI_OFF);
  unsigned* Mh         = ws + MH_OFF;
  float*    centers    = (float*)(ws + CT_OFF);
  float*    validA     = (float*)(ws + VA_OFF);
  float*    nvalid     = (float*)(ws + NV_OFF);
  unsigned* hist       = ws + HDR_WORDS;

  __shared__ int   semfS[64];
  __shared__ float cntS[64];
  if (tid < 64) {
    const int b = tid >> 4;
    const float cnt = centerSums[tid * 4 + 3];
    const float inv = 1.f / fmaxf(cnt, 1.f);
    centers[tid * 4 + 0] = centerSums[tid * 4 + 0] * inv;
    centers[tid * 4 + 1] = centerSums[tid * 4 + 1] * inv;
    centers[tid * 4 + 2] = centerSums[tid * 4 + 2] * inv;
    centers[tid * 4 + 3] = cnt;
    int sf = -1;
    if (cnt > 0.5f) sf = sem[b * NN + firstIdx[tid]];
    semfS[tid] = sf;
    cntS[tid]  = cnt;
  }
  __syncthreads();

  // tid enumerates seq = b*64 + c*16 + i
  const int b = tid >> 6;
  const int c = (tid >> 4) & 3;
  const int i = tid & 15;
  unsigned bulkFg = 0, totS = 0;
#pragma unroll
  for (int s = 0; s < 5; ++s)
    if (s != c + 1) bulkFg += Mh[b * 80 + s * 16 + i];
#pragma unroll
  for (int j = 0; j < 16; ++j) totS += Mh[b * 80 + (c + 1) * 16 + j];
  const unsigned bulkBg = (unsigned)NN - totS - bulkFg;   // error exactly 0
  unsigned* hb = hist + (size_t)tid * (size_t)nbins * 4u;
  hb[0] = bulkBg;                                          // bin 0, cntB
  hb[(size_t)(nbins - 1) * 4u + 1] = bulkFg;               // top bin, cntF
  ((float*)hb)[(size_t)(nbins - 1) * 4u + 3] = (float)bulkFg; // sumF (e==1)
  const float v = (cntS[b * 16 + i] > 0.5f && semfS[b * 16 + i] == c + 1) ? 1.f : 0.f;
  validA[tid] = v;
  atomicAdd(nvalid, v);
}

// Phase 3: per-point errors for sem-matching points, counting-sort histogram.
__global__ __launch_bounds__(256) void iou_errors(
    const float* __restrict__ pts, const int* __restrict__ tgt,
    const int* __restrict__ sem, const float* __restrict__ embd,
    const int* __restrict__ vrp, unsigned* __restrict__ ws, int nbins) {
  const int idx = blockIdx.x * 256 + threadIdx.x;  // over B*N
  const int b   = idx >> 17;                       // N = 2^17
  const int tid = threadIdx.x;
  const float* centers = (const float*)(ws + CT_OFF);
  unsigned*    hist    = ws + HDR_WORDS;

  __shared__ float cc[64];
  if (tid < 64) cc[tid] = centers[b * 64 + tid];
  __syncthreads();

  const int s = sem[idx];
  if (s < 1 || s > 4) return;                      // bulk-handled
  const float vr = (float)vrp[0];
  const float ex = embd[idx * 3 + 0] + pts[idx * 3 + 0] * vr;
  const float ey = embd[idx * 3 + 1] + pts[idx * 3 + 1] * vr;
  const float ez = embd[idx * 3 + 2] + pts[idx * 3 + 2] * vr;
  const int fgi  = tgt[idx];
  const int seqb = (b * 4 + (s - 1)) * 16;
  const float nbf = (float)nbins;
#pragma unroll 4
  for (int i = 0; i < 16; ++i) {
    const float dx = cc[i * 4 + 0] - ex;
    const float dy = cc[i * 4 + 1] - ey;
    const float dz = cc[i * 4 + 2] - ez;
    const float d2 = dx * dx + dy * dy + dz * dz;
    const float soft = __expf(-d2);                // exp(-d2/(2*0.5))
    const int fg = (i == fgi) ? 1 : 0;
    const float e = fg ? (1.f - soft) : soft;
    int bin = (int)(e * nbf);
    bin = min(max(bin, 0), nbins - 1);
    const size_t hb = ((size_t)(seqb + i) * (size_t)nbins + (size_t)bin) * 4u;
    atomicAdd(hist + hb + fg, 1u);
    atomicAdd((float*)hist + hb + 2 + fg, e);
  }
}

// Phase 4: one wave32 per sequence, descending-bin Lovasz scan.
// Within equal-error groups the Lovasz gradient telescopes, so
// contribution = meanE * (jac_incl - jac_excl).
__global__ __launch_bounds__(256) void iou_scan(unsigned* __restrict__ ws, int nbins) {
  const int tid  = threadIdx.x;
  const int lane = tid & 31;
  const int seq  = blockIdx.x * 8 + (tid >> 5);
  const float* validA  = (const float*)(ws + VA_OFF);
  const float* centers = (const float*)(ws + CT_OFF);
  float* lossSum = (float*)(ws + LS_OFF);
  const unsigned* hist = ws + HDR_WORDS;

  const float valid = validA[seq];
  if (valid == 0.f) return;                         // uniform per wave
  const int b = seq >> 6;
  const int i = seq & 15;
  const float gts = centers[(b * 16 + i) * 4 + 3];  // total fg count
  float runF = 0.f, runT = 0.f, loss = 0.f;
  const unsigned* hs = hist + (size_t)seq * (size_t)nbins * 4u;
  const int groups = nbins >> 5;
  for (int g = 0; g < groups; ++g) {
    const int bin = nbins - 1 - (g * 32 + lane);    // descending error order
    const unsigned* hp = hs + (size_t)bin * 4u;
    const unsigned cB = hp[0], cF = hp[1];
    const float sB = ((const float*)hp)[2];
    const float sF = ((const float*)hp)[3];
    const float nf = (float)cF;
    const float nt = (float)(cB + cF);
    // inclusive prefix across the 32-bin group (wave32 shfl scan)
    float exF = nf, exT = nt;
#pragma unroll
    for (int d = 1; d < 32; d <<= 1) {
      const float tF = __shfl_up(exF, d);
      const float tT = __shfl_up(exT, d);
      if (lane >= d) { exF += tF; exT += tT; }
    }
    const float F1 = runF + exF, T1 = runT + exT;   // inclusive of this bin
    const float F0 = F1 - nf,   T0 = T1 - nt;       // exclusive
    const float jac0 = 1.f - (gts - F0) / (gts + (T0 - F0));
    const float jac1 = 1.f - (gts - F1) / (gts + (T1 - F1));
    if (nt > 0.f) loss += (sB + sF) * (jac1 - jac0) / nt;
    runF += __shfl(exF, 31);
    runT += __shfl(exT, 31);
  }
#pragma unroll
  for (int off = 16; off > 0; off >>= 1) loss += __shfl_down(loss, off);
  if (lane == 0) atomicAdd(lossSum, loss * valid);
}

__global__ void iou_final(unsigned* __restrict__ ws, float* __restrict__ out) {
  const float nv = ((float*)ws)[NV_OFF];
  const float ls = ((float*)ws)[LS_OFF];
  out[0] = (nv > 0.f) ? ls / fmaxf(nv, 1.f) : 0.f;
}

extern "C" void kernel_launch(void* const* d_in, const int* in_sizes, int n_in,
                              void* d_out, int out_size, void* d_ws, size_t ws_size,
                              hipStream_t stream) {
  const float* pts  = (const float*)d_in[0];
  const int*   tgt  = (const int*)d_in[1];
  const int*   sem  = (const int*)d_in[2];
  const float* embd = (const float*)d_in[3];
  const int*   vrp  = (const int*)d_in[4];
  unsigned* ws  = (unsigned*)d_ws;
  float*    out = (float*)d_out;

  // choose histogram resolution from available scratch (16.8 MB at 4096 bins)
  const size_t wsWords = ws_size / 4;
  int nbins = 4096;
  while (nbins > 64 &&
         (size_t)HDR_WORDS + (size_t)NSEQ * (size_t)nbins * 4u > wsWords)
    nbins >>= 1;
  const int totalWords = HDR_WORDS + NSEQ * nbins * 4;

  iou_init<<<(totalWords + 255) / 256, 256, 0, stream>>>(ws, totalWords);
  dim3 g(BB, NN / CHUNK);
  iou_stats<<<g, 256, 0, stream>>>(pts, tgt, sem, embd, vrp, ws);
  iou_finalize<<<1, 256, 0, stream>>>(sem, ws, nbins);
  iou_errors<<<(BB * NN) / 256, 256, 0, stream>>>(pts, tgt, sem, embd, vrp, ws, nbins);
  iou_scan<<<NSEQ / 8, 256, 0, stream>>>(ws, nbins);
  iou_final<<<1, 1, 0, stream>>>(ws, out);
}